// Laplacian_80668075754214
// MI455X (gfx1250) — compile-verified
//
#include <hip/hip_runtime.h>

static constexpr int NV = 8192;
static constexpr int NF = 16384;

// ---------------------------------------------------------------------------
// Phase 1: zero-fill 256 MB via the CDNA5 async-store path.
// Each block: zero a 4 KB LDS staging buffer once, then every lane issues
// 16 x GLOBAL_STORE_ASYNC_FROM_LDS_B128 (ASYNCcnt-tracked, no VGPR data).
// Block covers 16 iters * 256 lanes * 16 B = 64 KB;  4096 blocks = 256 MB.
// ---------------------------------------------------------------------------
__global__ void __launch_bounds__(256) zero_async_kernel(float* __restrict__ out) {
    __shared__ float4 zbuf[256];
    zbuf[threadIdx.x] = make_float4(0.0f, 0.0f, 0.0f, 0.0f);

    // Raw LDS byte address of this lane's 16 B. Deriving it from the pointer
    // (ptrtoint) escapes zbuf so the zero-stores above cannot be DSE'd away;
    // generic LDS pointers carry the AS3 offset in their low 32 bits.
    unsigned lds_off = (unsigned)(uintptr_t)(void*)&zbuf[threadIdx.x];

    __syncthreads();   // all LDS zeroing complete (s_wait_dscnt + barrier)

    unsigned long long base = (unsigned long long)(uintptr_t)out
                            + (unsigned long long)blockIdx.x * 65536ull
                            + (unsigned long long)threadIdx.x * 16ull;

#define ZSTORE(OFF)                                                            \
    asm volatile("global_store_async_from_lds_b128 %0, %1, off offset:" #OFF   \
                 :: "v"(base), "v"(lds_off) : "memory")
    ZSTORE(0);     ZSTORE(4096);  ZSTORE(8192);  ZSTORE(12288);
    ZSTORE(16384); ZSTORE(20480); ZSTORE(24576); ZSTORE(28672);
    ZSTORE(32768); ZSTORE(36864); ZSTORE(40960); ZSTORE(45056);
    ZSTORE(49152); ZSTORE(53248); ZSTORE(57344); ZSTORE(61440);
#undef ZSTORE

    // Drain ASYNCcnt before wave exit (s_endpgm also implies wait-idle).
    asm volatile("s_wait_asynccnt 0x0" ::: "memory");
}

// ---------------------------------------------------------------------------
// Phase 2: per-face cotangent weights + scatter-add directly into L.
// ---------------------------------------------------------------------------
__device__ __forceinline__ float safe_norm(float dx, float dy, float dz) {
    float d2 = dx * dx + dy * dy + dz * dz;
    return d2 > 0.0f ? sqrtf(d2) : 0.0f;
}

__device__ __forceinline__ void fadd(float* p, float v) {
    // Relaxed, device-scope float atomic -> lowers to global_atomic_add_f32
    __hip_atomic_fetch_add(p, v, __ATOMIC_RELAXED, __HIP_MEMORY_SCOPE_AGENT);
}

__global__ void __launch_bounds__(256) face_kernel(const float* __restrict__ V,
                                                   const int*   __restrict__ F,
                                                   float*       __restrict__ L) {
    int f = blockIdx.x * 256 + threadIdx.x;
    if (f >= NF) return;

    // CDNA5 prefetch of the index stream (global_prefetch_b8)
    __builtin_prefetch(F + 3 * (size_t)f, 0, 0);

    int ia = F[3 * f + 0];
    int ib = F[3 * f + 1];
    int ic = F[3 * f + 2];

    float ax = V[3 * ia + 0], ay = V[3 * ia + 1], az = V[3 * ia + 2];
    float bx = V[3 * ib + 0], by = V[3 * ib + 1], bz = V[3 * ib + 2];
    float cx = V[3 * ic + 0], cy = V[3 * ic + 1], cz = V[3 * ic + 2];

    float l1 = safe_norm(bx - cx, by - cy, bz - cz); // |v2 - v3|
    float l2 = safe_norm(cx - ax, cy - ay, cz - az); // |v3 - v1|
    float l3 = safe_norm(ax - bx, ay - by, az - bz); // |v1 - v2|

    float sp = 0.5f * (l1 + l2 + l3);
    float s  = sp * (sp - l1) * (sp - l2) * (sp - l3);
    float A  = s > 0.0f ? 2.0f * sqrtf(s) : 1.0f;
    float inv4A = 1.0f / (4.0f * A);

    float q1 = l1 * l1, q2 = l2 * l2, q3 = l3 * l3;
    float c0 = (q2 + q3 - q1) * inv4A; // cot23: entry (ib, ic)
    float c1 = (q1 + q3 - q2) * inv4A; // cot31: entry (ic, ia)
    float c2 = (q1 + q2 - q3) * inv4A; // cot12: entry (ia, ib)

    size_t ra = (size_t)ia * NV;
    size_t rb = (size_t)ib * NV;
    size_t rc = (size_t)ic * NV;

    // Off-diagonal: L[r,c] -= v and L[c,r] -= v for each W0 entry.
    fadd(L + rb + ic, -c0);
    fadd(L + rc + ib, -c0);
    fadd(L + rc + ia, -c1);
    fadd(L + ra + ic, -c1);
    fadd(L + ra + ib, -c2);
    fadd(L + rb + ia, -c2);

    // Diagonal (row-sum) contributions, fused per vertex.
    fadd(L + ra + ia, c1 + c2);
    fadd(L + rb + ib, c0 + c2);
    fadd(L + rc + ic, c0 + c1);
}

extern "C" void kernel_launch(void* const* d_in, const int* in_sizes, int n_in,
                              void* d_out, int out_size, void* d_ws, size_t ws_size,
                              hipStream_t stream) {
    const float* V = (const float*)d_in[0];
    const int*   F = (const int*)d_in[1];
    float*       L = (float*)d_out;

    zero_async_kernel<<<4096, 256, 0, stream>>>(L);
    face_kernel<<<(NF + 255) / 256, 256, 0, stream>>>(V, F, L);
}